// MeanAggregator_64312840290798
// MI455X (gfx1250) — compile-verified
//
#include <hip/hip_runtime.h>
#include <hip/hip_bf16.h>
#include <stdint.h>

// MeanAggregator: out[r,:] = mean_s emb[neigh[r,s],:]
//   emb:   [100000, 128] fp32
//   neigh: [16384, 25]   int32
//   out:   [16384, 128]  fp32
//
// Bandwidth-bound gather-reduce. CDNA5 path: async global->LDS gather
// (GLOBAL_LOAD_ASYNC_TO_LDS_B128, ASYNCcnt) with per-wave double buffering.
// Indices are broadcast to SGPRs with v_readlane so the async loads use the
// scalar-base (SADDR) addressing form.

#define EMB_DIM   128
#define NSAMP     25
#define CHUNK     5                       // 25 = 5 chunks x 5 samples
#define NCHUNK    5
#define WAVES_PB  8                       // waves per block (256 threads)
#define ROW_BYTES (EMB_DIM * 4)           // 512 B per embedding row
#define CHUNK_BYTES (CHUNK * ROW_BYTES)   // 2560 B
#define WAVE_LDS_BYTES (2 * CHUNK_BYTES)  // 5120 B double buffer per wave

typedef __attribute__((ext_vector_type(4))) float v4f;
typedef __attribute__((ext_vector_type(4))) int   v4i;

// Pointer-to-v4i in global (AS1) and LDS (AS3) address spaces — the exact
// parameter types of the async-to-LDS builtin.
typedef __attribute__((address_space(1))) v4i* gv4i_p;
typedef __attribute__((address_space(3))) v4i* lv4i_p;

// ---- async global -> LDS copy of 16B per lane (512B per wave32) ------------
__device__ __forceinline__ void async_copy_b128(const float* gsrc, uint32_t lds_addr) {
#if __has_builtin(__builtin_amdgcn_global_load_async_to_lds_b128)
    __builtin_amdgcn_global_load_async_to_lds_b128(
        (gv4i_p)(uintptr_t)gsrc, (lv4i_p)(uintptr_t)lds_addr,
        /*offset=*/0, /*cpol=*/0);
#else
    unsigned long long ga = (unsigned long long)(uintptr_t)gsrc;
    asm volatile("global_load_async_to_lds_b128 %0, %1, off"
                 :: "v"(lds_addr), "v"(ga) : "memory");
#endif
}

__device__ __forceinline__ void wait_async_5() {
#if __has_builtin(__builtin_amdgcn_s_wait_asynccnt)
    __builtin_amdgcn_s_wait_asynccnt(5);
#else
    asm volatile("s_wait_asynccnt 5" ::: "memory");
#endif
}

__device__ __forceinline__ void wait_async_0() {
#if __has_builtin(__builtin_amdgcn_s_wait_asynccnt)
    __builtin_amdgcn_s_wait_asynccnt(0);
#else
    asm volatile("s_wait_asynccnt 0" ::: "memory");
#endif
}

__global__ __launch_bounds__(WAVES_PB * 32)
void MeanAggregator_64312840290798_kernel(const float* __restrict__ emb,
                                          const int*   __restrict__ neigh,
                                          float*       __restrict__ out,
                                          int batch) {
    __shared__ __align__(16) float smem[WAVES_PB * WAVE_LDS_BYTES / 4];

    const int lane = threadIdx.x & 31;
    const int wid  = threadIdx.x >> 5;
    const int row  = blockIdx.x * WAVES_PB + wid;
    if (row >= batch) return;

    // Wave-private LDS double buffer. Low 32 bits of the flat address of an
    // LDS object are the LDS byte offset (ISA 10.2 aperture mapping).
    float* wbase = smem + wid * (WAVE_LDS_BYTES / 4);
    const uint32_t lds_base = (uint32_t)(uintptr_t)(void*)wbase;

    // One coalesced load of this row's 25 neighbor ids (lanes 0..24), then
    // v_readlane broadcasts each to an SGPR inside the unrolled pipeline.
    int myidx = 0;
    if (lane < NSAMP) myidx = neigh[(size_t)row * NSAMP + lane];

    const uint32_t lane16 = (uint32_t)lane * 16u;
    const int      lane4  = lane * 4;

    auto issue_chunk = [&](int k, int p) {
        #pragma unroll
        for (int j = 0; j < CHUNK; ++j) {
            const int s   = k * CHUNK + j;
            const int idx = __builtin_amdgcn_readlane(myidx, s);  // SGPR
            const float* g = emb + (size_t)(unsigned)idx * EMB_DIM; // uniform base
            const uint32_t l = lds_base + (uint32_t)p * CHUNK_BYTES
                                        + (uint32_t)j * ROW_BYTES + lane16;
            async_copy_b128(g + lane4, l);   // saddr + lane voffset
        }
    };

    issue_chunk(0, 0);

    v4f acc = {0.0f, 0.0f, 0.0f, 0.0f};

    #pragma unroll
    for (int k = 0; k < NCHUNK; ++k) {
        if (k + 1 < NCHUNK) {
            issue_chunk(k + 1, (k + 1) & 1);   // prefetch next chunk
            wait_async_5();                    // in-order => chunk k landed
        } else {
            wait_async_0();                    // drain for the last chunk
        }
        asm volatile("" ::: "memory");         // keep LDS reads below the wait

        const float* cb = wbase + (k & 1) * (CHUNK_BYTES / 4);
        #pragma unroll
        for (int j = 0; j < CHUNK; ++j) {
            v4f v = *(const v4f*)(cb + j * EMB_DIM + lane4);     // ds_load_b128
            acc += v;
        }
    }

    acc *= (1.0f / (float)NSAMP);
    *(v4f*)(out + (size_t)row * EMB_DIM + lane4) = acc;          // global_store_b128
}

extern "C" void kernel_launch(void* const* d_in, const int* in_sizes, int n_in,
                              void* d_out, int out_size, void* d_ws, size_t ws_size,
                              hipStream_t stream) {
    const float* emb   = (const float*)d_in[0];
    const int*   neigh = (const int*)d_in[1];
    float*       out   = (float*)d_out;

    const int batch  = in_sizes[1] / NSAMP;                 // 16384
    const int blocks = (batch + WAVES_PB - 1) / WAVES_PB;   // 2048

    hipLaunchKernelGGL(MeanAggregator_64312840290798_kernel,
                       dim3(blocks), dim3(WAVES_PB * 32), 0, stream,
                       emb, neigh, out, batch);
}